// EWMA_model_56882546868814
// MI455X (gfx1250) — compile-verified
//
#include <hip/hip_runtime.h>
#include <math.h>

// Problem constants (match reference)
#define P      256     // assets
#define BLOCKD 20      // days per block
#define LAG    24
#define M      480     // LAG*BLOCKD train window
#define NT     512     // threads per workgroup (16 waves of 32)
#define NW     16      // waves
#define KC     16      // K-chunk staged in LDS per iteration
#define SPAD   257     // Sigma LDS row stride (floats) -> stride%64==1, conflict-free columns
#define APAD   264     // A-chunk LDS row stride: 2*264 % 64 == 16 -> half-wave groups hit disjoint banks
#define NTILE  136     // 16x16 tile count in lower triangle of 256x256
#define MAXT   9       // ceil(136/16) tiles per wave

typedef __attribute__((ext_vector_type(2))) float v2f;
typedef __attribute__((ext_vector_type(8))) float v8f;

__global__ __launch_bounds__(NT, 1)
void ewma_gmv_window_kernel(const float* __restrict__ Y,
                            const float* __restrict__ a_param,
                            float* __restrict__ e_out,
                            float* __restrict__ s_out,
                            int W)
{
    extern __shared__ float smem[];
    float* Sig  = smem;                 // P * SPAD      (263,168 B)
    float* As   = Sig  + P * SPAD;      // KC * APAD     ( 16,896 B)
    float* mean = As   + KC * APAD;     // P
    float* wts  = mean + P;             // M   (later holds sqrt(w))
    float* xv   = wts  + M;             // P
    float* rets = xv   + P;             // 32
    float* red  = rets + 32;            // NT

    const int tid    = threadIdx.x;
    const int lane   = tid & 31;
    const int waveId = tid >> 5;
    const int half   = lane >> 4;       // 0: lanes 0-15, 1: lanes 16-31
    const int m16    = lane & 15;
    const int win    = blockIdx.x;
    if (win >= W) return;

    // ---- EWMA weight statistics (closed form, each thread redundantly) ----
    const float a0    = a_param[0];
    const float S1    = (1.0f - powf(a0, (float)M)) / (1.0f - a0);       // sum a^t
    const float scale = (float)M / S1;                                   // w_t = scale*a^(M-1-t), sum w = M
    const float sw2   = scale * scale * (1.0f - powf(a0, 2.0f * (float)M)) / (1.0f - a0 * a0);
    const float denom = (float)M - sw2 / (float)M;                       // M*(1 - sum(w^2)/M^2)
    const float invdenom = 1.0f / denom;

    for (int t = tid; t < M; t += NT)
        wts[t] = scale * powf(a0, (float)(M - 1 - t));

    const long base = (long)win * BLOCKD * P;
    __syncthreads();

    // ---- weighted mean per column (rows split between the two thread halves) ----
    {
        int p  = tid & (P - 1);
        int h  = tid >> 8;                    // 0/1
        int t0 = h * (M / 2);
        float s = 0.0f;
        for (int t = t0; t < t0 + M / 2; ++t)
            s += wts[t] * Y[base + (long)t * P + p];
        red[tid] = s;
    }
    __syncthreads();
    if (tid < P) mean[tid] = (red[tid] + red[tid + P]) * (1.0f / (float)M);
    __syncthreads();
    // convert weights to sqrt(w) in place for Gram staging
    for (int t = tid; t < M; t += NT) wts[t] = sqrtf(wts[t]);
    __syncthreads();

    // ---- per-wave tile assignment over lower triangle (I >= J) ----
    int  IcA[MAXT], JcA[MAXT];
    bool valid[MAXT];
    #pragma unroll
    for (int ti = 0; ti < MAXT; ++ti) {
        int t = waveId + NW * ti;
        valid[ti] = (t < NTILE);
        int tt = valid[ti] ? t : 0;
        int I = (int)((sqrtf(8.0f * (float)tt + 1.0f) - 1.0f) * 0.5f);
        while ((I + 1) * (I + 2) / 2 <= tt) ++I;
        while (I * (I + 1) / 2 > tt) --I;
        int J = tt - I * (I + 1) / 2;
        IcA[ti] = I * 16;
        JcA[ti] = J * 16;
    }

    v8f acc[MAXT];
    #pragma unroll
    for (int ti = 0; ti < MAXT; ++ti) acc[ti] = (v8f){};

    // ---- Sigma = (sqrt(w)(Y-mean))^T (sqrt(w)(Y-mean)) via f32 WMMA, K-chunked ----
    for (int kc = 0; kc < M; kc += KC) {
        for (int i = tid; i < KC * P; i += NT) {         // stage chunk, coalesced
            int t = i >> 8;
            int p = i & (P - 1);
            As[t * APAD + p] = wts[kc + t] * (Y[base + (long)(kc + t) * P + p] - mean[p]);
        }
        __syncthreads();
        #pragma unroll
        for (int ti = 0; ti < MAXT; ++ti) {
            if (valid[ti]) {                              // wave-uniform: EXEC stays all-ones
                const float* colI = As + IcA[ti] + m16;
                const float* colJ = As + JcA[ti] + m16;
                #pragma unroll
                for (int kk = 0; kk < KC; kk += 4) {
                    int r0 = kk + 2 * half;               // ISA fp32 A/B frag layout: K 0,1 | 2,3
                    v2f av, bv;
                    av[0] = colI[(r0    ) * APAD];
                    av[1] = colI[(r0 + 1) * APAD];
                    bv[0] = colJ[(r0    ) * APAD];
                    bv[1] = colJ[(r0 + 1) * APAD];
                    acc[ti] = __builtin_amdgcn_wmma_f32_16x16x4_f32(
                        false, av, false, bv, (short)0, acc[ti], false, false);
                }
            }
        }
        __syncthreads();
    }

    // ---- write lower-triangle Sigma tiles to LDS (D layout: M = v + 8*half, N = m16) ----
    #pragma unroll
    for (int ti = 0; ti < MAXT; ++ti) {
        if (valid[ti]) {
            #pragma unroll
            for (int v = 0; v < 8; ++v) {
                int row = IcA[ti] + v + 8 * half;
                Sig[row * SPAD + JcA[ti] + m16] = acc[ti][v] * invdenom;
            }
        }
    }
    __syncthreads();

    // ---- in-LDS Cholesky (lower triangle, right-looking) ----
    for (int k = 0; k < P; ++k) {
        if (tid == 0) Sig[k * SPAD + k] = sqrtf(Sig[k * SPAD + k]);
        __syncthreads();
        float dinv = 1.0f / Sig[k * SPAD + k];
        for (int i = k + 1 + tid; i < P; i += NT)
            Sig[i * SPAD + k] *= dinv;
        __syncthreads();
        int n = P - 1 - k;
        int T = n * (n + 1) / 2;                          // trailing lower triangle incl diag
        for (int idx = tid; idx < T; idx += NT) {
            int a = (int)((sqrtf(8.0f * (float)idx + 1.0f) - 1.0f) * 0.5f);
            while ((a + 1) * (a + 2) / 2 <= idx) ++a;
            while (a * (a + 1) / 2 > idx) --a;
            int b = idx - a * (a + 1) / 2;
            int i = k + 1 + a;
            int j = k + 1 + b;
            Sig[i * SPAD + j] -= Sig[i * SPAD + k] * Sig[j * SPAD + k];
        }
        __syncthreads();
    }

    // ---- solve L L^T x = 1 ----
    if (tid < P) xv[tid] = 1.0f;
    __syncthreads();
    for (int k = 0; k < P; ++k) {                         // forward: L y = 1
        if (tid == 0) xv[k] /= Sig[k * SPAD + k];
        __syncthreads();
        float xk = xv[k];
        int i = k + 1 + tid;
        if (i < P) xv[i] -= Sig[i * SPAD + k] * xk;
        __syncthreads();
    }
    for (int k = P - 1; k >= 0; --k) {                    // backward: L^T x = y
        if (tid == 0) xv[k] /= Sig[k * SPAD + k];
        __syncthreads();
        float xk = xv[k];
        if (tid < k) xv[tid] -= Sig[k * SPAD + tid] * xk;
        __syncthreads();
    }

    // ---- sum(x) for GMV normalization ----
    red[tid] = (tid < P) ? xv[tid] : 0.0f;
    __syncthreads();
    for (int off = NT / 2; off > 0; off >>= 1) {
        if (tid < off) red[tid] += red[tid + off];
        __syncthreads();
    }
    float invxs = 1.0f / red[0];

    // ---- out-of-sample returns over 20 test days (one row per wave, shuffle reduce) ----
    for (int t = waveId; t < BLOCKD; t += NW) {
        const float* yrow = Y + base + (long)(M + t) * P;
        float r = 0.0f;
        for (int q = lane; q < P; q += 32)
            r += yrow[q] * xv[q];
        for (int off = 16; off > 0; off >>= 1)
            r += __shfl_down(r, off, 32);                 // wave32
        if (lane == 0) rets[t] = r * invxs;
    }
    __syncthreads();
    if (tid == 0) {
        float e = 0.0f;
        for (int t = 0; t < BLOCKD; ++t) e += rets[t];
        float mr = e / (float)BLOCKD;
        float s = 0.0f;
        for (int t = 0; t < BLOCKD; ++t) { float d = rets[t] - mr; s += d * d; }
        s /= (float)BLOCKD;                               // ddof=0, matches jnp.var
        e_out[win] = e;
        s_out[win] = s;
    }
}

__global__ void ewma_finalize_kernel(const float* __restrict__ e,
                                     const float* __restrict__ s,
                                     float* __restrict__ out, int W)
{
    __shared__ float re[NT], rs[NT];
    int tid = threadIdx.x;
    float se = 0.0f, ss = 0.0f;
    for (int i = tid; i < W; i += NT) { se += e[i]; ss += s[i]; }
    re[tid] = se; rs[tid] = ss;
    __syncthreads();
    for (int off = NT / 2; off > 0; off >>= 1) {
        if (tid < off) { re[tid] += re[tid + off]; rs[tid] += rs[tid + off]; }
        __syncthreads();
    }
    if (tid == 0) {
        float mean_s = rs[0] / (float)W;
        float mean_e = re[0] / (float)W;
        float vol = sqrtf(mean_s * 252.0f);
        float mu  = mean_e / (float)BLOCKD * 252.0f;
        out[0] = vol;
        out[1] = mu;
        out[2] = mu / vol;
    }
}

extern "C" void kernel_launch(void* const* d_in, const int* in_sizes, int n_in,
                              void* d_out, int out_size, void* d_ws, size_t ws_size,
                              hipStream_t stream) {
    const float* Y = (const float*)d_in[0];
    const float* a = (const float*)d_in[1];
    float* out = (float*)d_out;

    int n = in_sizes[0] / P;          // days
    int W = n / BLOCKD - LAG;         // rolling windows (512 for the reference shapes)

    float* e = (float*)d_ws;          // W floats
    float* s = e + W;                 // W floats  (8 KB total worst case)

    size_t smemBytes = (size_t)(P * SPAD + KC * APAD + P + M + P + 32 + NT) * sizeof(float);

    ewma_gmv_window_kernel<<<W, NT, smemBytes, stream>>>(Y, a, e, s, W);
    ewma_finalize_kernel<<<1, NT, 0, stream>>>(e, s, out, W);
}